// ARMA_79053168050941
// MI455X (gfx1250) — compile-verified
//
#include <hip/hip_runtime.h>
#include <hip/hip_bf16.h>

// ---------------------------------------------------------------------------
// ARMA graph conv (3 layers), F = 256, fp32 throughout.
// GEMMs: V_WMMA_F32_16X16X4_F32 (fp32 matrix cores, exact precision).
//   - 64 M-rows per block, 8 waves; each wave: 4 stacked 16x16 tiles so one
//     B fragment feeds 4 WMMAs (4x less W traffic, 4x WMMA density).
// Edge scatter: one wave32 per edge, float4 gathers + global f32 atomics
// (acc fits in 192MB L2 so atomics resolve on-chip).
// ---------------------------------------------------------------------------

#define FDIM 256
#define LDS_PAD 4   // row stride 260 floats: stride-4 bank spread, 16B aligned
#define MTILES 4    // 16-row tiles stacked per wave
#define MROWS  (MTILES * 16)  // 64 rows per block

typedef __attribute__((ext_vector_type(2))) float v2f;
typedef __attribute__((ext_vector_type(8))) float v8f;

// ---------------- small helper kernels ----------------

__global__ void zero_f32(float* __restrict__ p, int n) {
    int i = blockIdx.x * blockDim.x + threadIdx.x;
    if (i < n) p[i] = 0.0f;
}

__global__ void deg_scatter(const float* __restrict__ w,
                            const int* __restrict__ dst,
                            float* __restrict__ deg, int E) {
    int e = blockIdx.x * blockDim.x + threadIdx.x;
    if (e < E) atomicAdd(&deg[dst[e]], w[e]);
}

// deg -> dinv in place
__global__ void deg_to_dinv(float* __restrict__ deg, int n) {
    int i = blockIdx.x * blockDim.x + threadIdx.x;
    if (i < n) {
        float d = deg[i];
        deg[i] = (d > 0.0f) ? rsqrtf(d) : 0.0f;
    }
}

__global__ void edge_norm(const float* __restrict__ dinv,
                          const float* __restrict__ w,
                          const int* __restrict__ src,
                          const int* __restrict__ dst,
                          float* __restrict__ norm, int E) {
    int e = blockIdx.x * blockDim.x + threadIdx.x;
    if (e < E) norm[e] = dinv[src[e]] * w[e] * dinv[dst[e]];
}

__global__ void relu_inplace(float* __restrict__ p, int n) {
    int i = blockIdx.x * blockDim.x + threadIdx.x;
    if (i < n) p[i] = fmaxf(p[i], 0.0f);
}

// ---------------- WMMA fp32 GEMM: Out = X @ W (+ bias) ----------------
// X: [nrows, 256], W: [256, 256] row-major, Out: [nrows, 256]
// block = 256 threads = 8 waves. Block covers 64 rows x 128 cols:
// wave w owns cols [128*by + 16*w, +16) and 4 stacked 16-row tiles.
__global__ void __launch_bounds__(256)
gemm_wmma_f32(const float* __restrict__ X, const float* __restrict__ W,
              const float* __restrict__ bias, float* __restrict__ Out,
              int nrows) {
    __shared__ float xs[MROWS][FDIM + LDS_PAD];

    const int tid  = threadIdx.x;
    const int row0 = blockIdx.x * MROWS;

    // cooperative staging: 64 rows x 64 float4 = 4096 float4, 16 per thread
    #pragma unroll
    for (int j = 0; j < 16; ++j) {
        int i   = j * 256 + tid;      // linear float4 index
        int r   = i >> 6;             // row within tile (64 float4 per row)
        int c4  = i & 63;             // float4 column
        int gr  = row0 + r;
        float4 v = make_float4(0.f, 0.f, 0.f, 0.f);
        if (gr < nrows) v = *(const float4*)(X + (size_t)gr * FDIM + c4 * 4);
        *(float4*)(&xs[r][c4 * 4]) = v;
    }
    __syncthreads();

    const int wave = tid >> 5;
    const int lane = tid & 31;
    const int half = lane >> 4;      // 0: lanes 0-15, 1: lanes 16-31
    const int lm   = lane & 15;
    const int n    = blockIdx.y * 128 + wave * 16 + lm;  // output column

    v8f c[MTILES];
    #pragma unroll
    for (int t = 0; t < MTILES; ++t) c[t] = v8f{0.f,0.f,0.f,0.f,0.f,0.f,0.f,0.f};

    // K loop: one B fragment per k-step feeds 4 WMMAs (one per M tile)
    #pragma unroll 4
    for (int k0 = 0; k0 < FDIM; k0 += 4) {
        const int k = k0 + 2 * half;
        // B 4x16 frag: b[v] = B[k0 + 2*half + v][n]
        v2f b;
        b.x = W[(size_t)k * FDIM + n];
        b.y = W[(size_t)(k + 1) * FDIM + n];
        #pragma unroll
        for (int t = 0; t < MTILES; ++t) {
            // A 16x4 frag: a[v] = A[m][k0 + 2*half + v]
            v2f a;
            a.x = xs[t * 16 + lm][k];
            a.y = xs[t * 16 + lm][k + 1];
            c[t] = __builtin_amdgcn_wmma_f32_16x16x4_f32(
                       false, a, false, b, (short)0, c[t], false, false);
        }
    }

    const float bv = bias ? bias[n] : 0.0f;
    // C/D layout: VGPR v, lanes 0-15 -> M=v, lanes 16-31 -> M=v+8
    if (row0 + MROWS <= nrows) {
        // full tile: unguarded stores
        #pragma unroll
        for (int t = 0; t < MTILES; ++t) {
            const size_t rbase = (size_t)(row0 + t * 16 + 8 * half);
            #pragma unroll
            for (int v = 0; v < 8; ++v)
                Out[(rbase + v) * FDIM + n] = c[t][v] + bv;
        }
    } else {
        #pragma unroll
        for (int t = 0; t < MTILES; ++t) {
            #pragma unroll
            for (int v = 0; v < 8; ++v) {
                int r = row0 + t * 16 + v + 8 * half;
                if (r < nrows) Out[(size_t)r * FDIM + n] = c[t][v] + bv;
            }
        }
    }
}

// ---------------- edge scatter: acc[dst] += norm * h[src] ----------------
// one wave32 per edge; 256 floats = 2 x float4 per lane; f32 atomics to L2.
__global__ void __launch_bounds__(256)
edge_scatter_add(const float* __restrict__ h, const float* __restrict__ norm,
                 const int* __restrict__ src, const int* __restrict__ dst,
                 float* __restrict__ acc, int E) {
    const int wavesPerBlock = blockDim.x >> 5;
    const int e    = blockIdx.x * wavesPerBlock + (threadIdx.x >> 5);
    const int lane = threadIdx.x & 31;
    if (e >= E) return;

    const int   s  = src[e];
    const int   d  = dst[e];
    const float nv = norm[e];

    const float4* hp = (const float4*)(h + (size_t)s * FDIM);
    float*        ap = acc + (size_t)d * FDIM;

    float4 v0 = hp[lane];        // cols 4*lane   .. 4*lane+3
    float4 v1 = hp[lane + 32];   // cols 128+4*lane ..

    int c0 = 4 * lane;
    atomicAdd(ap + c0 + 0, nv * v0.x);
    atomicAdd(ap + c0 + 1, nv * v0.y);
    atomicAdd(ap + c0 + 2, nv * v0.z);
    atomicAdd(ap + c0 + 3, nv * v0.w);
    int c1 = 128 + 4 * lane;
    atomicAdd(ap + c1 + 0, nv * v1.x);
    atomicAdd(ap + c1 + 1, nv * v1.y);
    atomicAdd(ap + c1 + 2, nv * v1.z);
    atomicAdd(ap + c1 + 3, nv * v1.w);
}

// ---------------------------------------------------------------------------

static inline size_t align_elems(size_t n) { return (n + 63) & ~(size_t)63; }

extern "C" void kernel_launch(void* const* d_in, const int* in_sizes, int n_in,
                              void* d_out, int out_size, void* d_ws, size_t ws_size,
                              hipStream_t stream) {
    // inputs: x, edge_index, edge_attr, W_init, W_root, bias
    const float* x      = (const float*)d_in[0];
    const int*   eidx   = (const int*)d_in[1];
    const float* eattr  = (const float*)d_in[2];
    const float* W_init = (const float*)d_in[3];
    const float* W_root = (const float*)d_in[4];
    const float* bias   = (const float*)d_in[5];

    const int NF = in_sizes[0];          // N * 256
    const int Nn = NF / FDIM;            // 50000
    const int E  = in_sizes[2];          // 800000
    const int L  = in_sizes[3] / (FDIM * FDIM);  // 3

    const int* srcv = eidx;
    const int* dstv = eidx + E;

    // workspace layout (floats)
    float* ws   = (float*)d_ws;
    float* dinv = ws;                                   // N   (deg, then dinv)
    float* norm = dinv + align_elems(Nn);               // E
    float* hbuf = norm + align_elems(E);                // N*F
    float* bufA = hbuf + align_elems((size_t)NF);       // N*F
    float* bufB = bufA + align_elems((size_t)NF);       // N*F

    const int T = 256;
    // ---- edge-constant preprocessing (once) ----
    zero_f32<<<(Nn + T - 1) / T, T, 0, stream>>>(dinv, Nn);
    deg_scatter<<<(E + T - 1) / T, T, 0, stream>>>(eattr, dstv, dinv, E);
    deg_to_dinv<<<(Nn + T - 1) / T, T, 0, stream>>>(dinv, Nn);
    edge_norm<<<(E + T - 1) / T, T, 0, stream>>>(dinv, eattr, srcv, dstv, norm, E);

    const dim3 gemmGrid((Nn + MROWS - 1) / MROWS, FDIM / 128);  // (782, 2)
    const int  scatterBlocks = (E + 7) / 8;                     // 8 waves/block

    const float* xin = x;
    for (int l = 0; l < L; ++l) {
        float* acc = (l == L - 1) ? (float*)d_out : (l == 0 ? bufA : bufB);
        const float* Wi = W_init + (size_t)l * FDIM * FDIM;
        const float* Wr = W_root + (size_t)l * FDIM * FDIM;
        const float* bl = bias + (size_t)l * FDIM;

        // h = x @ Wi
        gemm_wmma_f32<<<gemmGrid, T, 0, stream>>>(xin, Wi, nullptr, hbuf, Nn);
        // acc = x @ Wr + b
        gemm_wmma_f32<<<gemmGrid, T, 0, stream>>>(xin, Wr, bl, acc, Nn);
        // acc += segment_sum(norm * h[src] -> dst)
        edge_scatter_add<<<scatterBlocks, T, 0, stream>>>(hbuf, norm, srcv, dstv, acc, E);
        // acc = relu(acc)
        relu_inplace<<<(NF + T - 1) / T, T, 0, stream>>>(acc, NF);

        xin = acc;
    }
}